// GaussianModel_78554951844281
// MI455X (gfx1250) — compile-verified
//
#include <hip/hip_runtime.h>
#include <hip/hip_bf16.h>

#define SH_C0 0.28209479177387814f
#define SH_C1 0.4886025119029199f

typedef __attribute__((ext_vector_type(2))) float v2f;
typedef __attribute__((ext_vector_type(4))) float v4f;
typedef __attribute__((ext_vector_type(8))) float v8f;

__device__ __forceinline__ float sigmoidf(float x) {
    return 1.0f / (1.0f + expf(-x));
}

// ---------------------------------------------------------------------------
// Kernel 1: pure streaming elementwise work (xyz passthrough, SH->rgb, opacity)
// One thread per gaussian; non-temporal since every byte is touched once.
// ---------------------------------------------------------------------------
__global__ __launch_bounds__(256) void gs_elementwise(
    const float* __restrict__ view_dirs,     // (N,3)
    const float* __restrict__ xyz,           // (N,3)
    const float* __restrict__ opacity_logit, // (N,1)
    const float* __restrict__ sh,            // (N,3,4)
    float* __restrict__ out_xyz,             // N*3
    float* __restrict__ out_rgb,             // N*3
    float* __restrict__ out_opa,             // N
    int N)
{
    int g = blockIdx.x * blockDim.x + threadIdx.x;
    if (g >= N) return;
    size_t g3 = (size_t)g * 3;

    float px = __builtin_nontemporal_load(xyz + g3 + 0);
    float py = __builtin_nontemporal_load(xyz + g3 + 1);
    float pz = __builtin_nontemporal_load(xyz + g3 + 2);
    __builtin_nontemporal_store(px, out_xyz + g3 + 0);
    __builtin_nontemporal_store(py, out_xyz + g3 + 1);
    __builtin_nontemporal_store(pz, out_xyz + g3 + 2);

    float dx = __builtin_nontemporal_load(view_dirs + g3 + 0);
    float dy = __builtin_nontemporal_load(view_dirs + g3 + 1);
    float dz = __builtin_nontemporal_load(view_dirs + g3 + 2);

    const v4f* shv = (const v4f*)(sh + (size_t)g * 12); // 48B/row, 16B aligned
#pragma unroll
    for (int c = 0; c < 3; ++c) {
        v4f s = __builtin_nontemporal_load(shv + c);
        float v = SH_C0 * s.x - SH_C1 * dy * s.y + SH_C1 * dz * s.z - SH_C1 * dx * s.w;
        __builtin_nontemporal_store(sigmoidf(v), out_rgb + g3 + c);
    }

    float ol = __builtin_nontemporal_load(opacity_logit + g);
    __builtin_nontemporal_store(sigmoidf(ol), out_opa + g);
}

// ---------------------------------------------------------------------------
// Kernel 2: cov3d = M * M^T via V_WMMA_F32_16X16X4_F32, 5 gaussians per WMMA.
// Gaussian g (0..4) occupies rows 3g..3g+2 of A (row 15 zero). For the
// symmetric product A*A^T the B operand's column-per-lane layout is
// register-identical to A's row-per-lane layout, so A is passed twice.
// D's diagonal 3x3 blocks are the covariances.
//
// A (f32 16x4): lanes 0-15 = row m, {v0=K0, v1=K1}; lanes 16-31 = row m-16,
// {v0=K2, v1=K3=0}.  D (f32 16x16): VGPR v -> (M=v, N=lane) low half,
// (M=v+8, N=lane-16) high half.  EXEC all-ones at every WMMA (the staging /
// scatter `if`s re-converge before it, and all launched blocks are full).
// ---------------------------------------------------------------------------
#define GS_PER_WMMA  5
#define WMMA_ITERS   8
#define GS_PER_WAVE  (GS_PER_WMMA * WMMA_ITERS)  // 40
#define GS_PER_BLOCK (8 * GS_PER_WAVE)           // 320 (256 threads = 8 waves)

__global__ __launch_bounds__(256) void gs_cov_wmma(
    const float* __restrict__ scale_log, // (N,3)
    const float* __restrict__ quat,      // (N,4)
    float* __restrict__ out_cov,         // N*9
    int N)
{
    const int lane = threadIdx.x & 31;
    const int wave = threadIdx.x >> 5;
    const int hi   = lane >> 4;   // 16-lane half
    const int nl   = lane & 15;   // slot within half

    const int wbase = blockIdx.x * GS_PER_BLOCK + wave * GS_PER_WAVE;

    // Per-lane constants for staging/scatter.
    const int gl = nl / 3;          // local gaussian 0..4 (nl==15 unused)
    const int iRow = nl - 3 * gl;   // row of M within the 3x3 block

#pragma unroll
    for (int it = 0; it < WMMA_ITERS; ++it) {
        const int gbase = wbase + it * GS_PER_WMMA;

        // ---- stage A: this lane owns row iRow of local gaussian gl ----
        float a0 = 0.0f, a1 = 0.0f;
        int gg = gbase + gl;
        if (nl < 15 && gg < N) {
            v4f q = ((const v4f*)quat)[gg];
            float nrm = sqrtf(q.x * q.x + q.y * q.y + q.z * q.z + q.w * q.w);
            float inv = 1.0f / fmaxf(nrm, 1e-12f);
            float r = q.x * inv, x = q.y * inv, y = q.z * inv, z = q.w * inv;

            size_t s3 = (size_t)gg * 3;
            float s0 = expf(scale_log[s3 + 0]);
            float s1 = expf(scale_log[s3 + 1]);
            float s2 = expf(scale_log[s3 + 2]);

            float R0, R1, R2;
            if (iRow == 0)      { R0 = 1.f - 2.f*(y*y + z*z); R1 = 2.f*(x*y - r*z); R2 = 2.f*(x*z + r*y); }
            else if (iRow == 1) { R0 = 2.f*(x*y + r*z); R1 = 1.f - 2.f*(x*x + z*z); R2 = 2.f*(y*z - r*x); }
            else                { R0 = 2.f*(x*z - r*y); R1 = 2.f*(y*z + r*x); R2 = 1.f - 2.f*(x*x + y*y); }

            float m0 = R0 * s0, m1 = R1 * s1, m2 = R2 * s2;
            a0 = hi ? m2 : m0;   // low half: K0 ; high half: K2
            a1 = hi ? 0.f : m1;  // low half: K1 ; high half: K3 = 0
        }
        v2f a = {a0, a1};

        v8f c = {};
        c = __builtin_amdgcn_wmma_f32_16x16x4_f32(
                /*neg_a=*/false, a, /*neg_b=*/false, a,
                /*c_mod=*/(short)0, c, /*reuse_a=*/false, /*reuse_b=*/false);

        // ---- scatter diagonal 3x3 blocks: lane owns D column n = nl ----
        if (nl < 15) {
            int go = gbase + gl;
            if (go < N) {
#pragma unroll
                for (int ii = 0; ii < 3; ++ii) {
                    int m  = 3 * gl + ii;   // row in D
                    int mh = m >> 3;        // which half holds this row
                    if (mh == hi) {
                        __builtin_nontemporal_store(
                            c[m & 7], out_cov + (size_t)go * 9 + ii * 3 + nl - 3 * gl);
                    }
                }
            }
        }
    }
}

extern "C" void kernel_launch(void* const* d_in, const int* in_sizes, int n_in,
                              void* d_out, int out_size, void* d_ws, size_t ws_size,
                              hipStream_t stream) {
    const float* view_dirs     = (const float*)d_in[0];
    const float* xyz           = (const float*)d_in[1];
    const float* scale_log     = (const float*)d_in[2];
    const float* rot_quat      = (const float*)d_in[3];
    const float* opacity_logit = (const float*)d_in[4];
    const float* sh_coeffs     = (const float*)d_in[5];

    int N = in_sizes[1] / 3;   // xyz is (N,3)

    float* out     = (float*)d_out;
    float* out_xyz = out;                      // [0, 3N)
    float* out_cov = out + (size_t)N * 3;      // [3N, 12N)
    float* out_rgb = out + (size_t)N * 12;     // [12N, 15N)
    float* out_opa = out + (size_t)N * 15;     // [15N, 16N)

    int blocks_ew = (N + 255) / 256;
    gs_elementwise<<<blocks_ew, 256, 0, stream>>>(
        view_dirs, xyz, opacity_logit, sh_coeffs,
        out_xyz, out_rgb, out_opa, N);

    int blocks_cov = (N + GS_PER_BLOCK - 1) / GS_PER_BLOCK;  // 6250 for N=2M
    gs_cov_wmma<<<blocks_cov, 256, 0, stream>>>(
        scale_log, rot_quat, out_cov, N);
}